// GRUCell2_52175262712016
// MI455X (gfx1250) — compile-verified
//
#include <hip/hip_runtime.h>
#include <math.h>

// Problem constants (match reference)
#define B_ 8
#define P_ 65536
#define C_ 32768
#define F_ 32
#define E_ 524288

typedef __attribute__((ext_vector_type(2))) float v2f;
typedef __attribute__((ext_vector_type(8))) float v8f;

__device__ __forceinline__ int lower_bound_i32(const int* __restrict__ a, int n, int key) {
    int lo = 0, hi = n;
    while (lo < hi) {
        int mid = (lo + hi) >> 1;
        if (a[mid] < key) lo = mid + 1; else hi = mid;
    }
    return lo;
}

__device__ __forceinline__ float sigmoid_f(float x) {
    return 1.0f / (1.0f + __expf(-x));
}

// One wave32 handles a 16-row tile of the [B*C, F] problem:
//   - segment-sum 16 channels of x (lane = feature, coalesced 128B loads/edge)
//   - stage 16x32 x-tile in LDS
//   - 96x v_wmma_f32_16x16x4_f32 over K=32 for gates r,z,n (ih and hh GEMMs)
//   - gate math + final blend in C/D fragment layout
__launch_bounds__(256)
__global__ void gru_edge_fused_kernel(const float* __restrict__ pathf,   // [B,P,F]
                                      const float* __restrict__ chanf,   // [B,C,F] (= h)
                                      const float* __restrict__ W_ih,    // [96,32]
                                      const float* __restrict__ W_hh,    // [96,32]
                                      const float* __restrict__ b_ih,    // [96]
                                      const float* __restrict__ b_hh,    // [96]
                                      const int*   __restrict__ edge_path,    // [E]
                                      const int*   __restrict__ edge_channel, // [E] sorted
                                      float* __restrict__ out)           // [B,C,F]
{
    __shared__ float lds_w[2 * 96 * 32];   // W_ih then W_hh (24 KB)
    __shared__ float lds_x[8 * 16 * 32];   // per-wave 16x32 x staging (16 KB)

    const int tid = threadIdx.x;

    // Cooperative weight preload into LDS (once per workgroup)
    for (int i = tid; i < 96 * 32; i += 256) {
        lds_w[i]           = W_ih[i];
        lds_w[96 * 32 + i] = W_hh[i];
    }

    const int wave = tid >> 5;
    const int lane = tid & 31;
    const int tile = blockIdx.x * 8 + wave;     // 16-row tile id, rows = b*C + c
    const int row0 = tile * 16;
    const int b    = row0 / C_;                 // constant across tile (16 | C)
    const int c0   = row0 % C_;

    float* xs = &lds_x[wave * (16 * 32)];

    // ---- Stage 1: deterministic segment-sum (edge_channel is sorted) ----
    // lane == feature index; per edge: one coalesced 128B wave load
    const float* pb = pathf + (size_t)b * P_ * F_;
    int lo = lower_bound_i32(edge_channel, E_, c0);
    for (int r = 0; r < 16; ++r) {
        const int hi = lower_bound_i32(edge_channel, E_, c0 + r + 1);
        float acc = 0.0f;
        for (int e = lo; e < hi; ++e) {
            const int p = edge_path[e];                  // wave-uniform
            acc += pb[(size_t)p * F_ + lane];
        }
        xs[r * 32 + lane] = acc;
        lo = hi;
    }
    __syncthreads();   // weights visible + x staging ordered

    // ---- Stage 2: A fragments (32-bit A 16x4 layout, 8 K-steps of 4) ----
    // lane l: row = l%16, K base = 4*kb + 2*(l/16)  -> 8B loads
    const int arow = lane & 15;
    const int ak   = (lane >> 4) << 1;           // 0 or 2
    const int bcol = lane & 15;                  // N index for B/C/D frags

    const float* hbase = chanf + (size_t)(b * C_ + c0) * F_;

    v2f ax[8], ah[8];
#pragma unroll
    for (int kb = 0; kb < 8; ++kb) {
        const int k0 = kb * 4 + ak;
        ax[kb] = *(const v2f*)&xs[arow * 32 + k0];                 // ds_load_b64
        ah[kb] = *(const v2f*)&hbase[(size_t)arow * 32 + k0];      // global b64
    }

    // B fragment: col-tile jt (16 cols of the 96), K-step kb, weight wsel
    auto bfrag = [&](int wsel, int jt, int kb) -> v2f {
        const int j  = jt * 16 + bcol;
        const int k0 = kb * 4 + ak;
        return *(const v2f*)&lds_w[wsel * (96 * 32) + j * 32 + k0]; // ds_load_b64
    };

    // ---- Stage 3: WMMA accumulation (f32 matrix core, full precision) ----
    v8f accR[2], accZ[2], accXN[2], accHN[2];
#pragma unroll
    for (int ft = 0; ft < 2; ++ft) {
        v8f r = {}, z = {}, xn = {}, hn = {};
#pragma unroll
        for (int kb = 0; kb < 8; ++kb) {
            // x-side GEMM (W_ih): gates r, z, n
            r  = __builtin_amdgcn_wmma_f32_16x16x4_f32(false, ax[kb], false, bfrag(0, 0 + ft, kb), (short)0, r,  false, false);
            z  = __builtin_amdgcn_wmma_f32_16x16x4_f32(false, ax[kb], false, bfrag(0, 2 + ft, kb), (short)0, z,  false, false);
            xn = __builtin_amdgcn_wmma_f32_16x16x4_f32(false, ax[kb], false, bfrag(0, 4 + ft, kb), (short)0, xn, false, false);
            // h-side GEMM (W_hh): r,z chain into same accumulators; n kept separate
            r  = __builtin_amdgcn_wmma_f32_16x16x4_f32(false, ah[kb], false, bfrag(1, 0 + ft, kb), (short)0, r,  false, false);
            z  = __builtin_amdgcn_wmma_f32_16x16x4_f32(false, ah[kb], false, bfrag(1, 2 + ft, kb), (short)0, z,  false, false);
            hn = __builtin_amdgcn_wmma_f32_16x16x4_f32(false, ah[kb], false, bfrag(1, 4 + ft, kb), (short)0, hn, false, false);
        }
        accR[ft] = r; accZ[ft] = z; accXN[ft] = xn; accHN[ft] = hn;
    }

    // ---- Stage 4: gates + blend in C/D layout (VGPR i: M = i + 8*(lane/16), N = lane%16)
    float* ob = out + (size_t)(b * C_ + c0) * F_;
    const int mhi = (lane >> 4) << 3;            // 0 or 8
#pragma unroll
    for (int ft = 0; ft < 2; ++ft) {
        const int j   = ft * 16 + bcol;          // output feature
        const float br  = b_ih[j]      + b_hh[j];
        const float bz  = b_ih[32 + j] + b_hh[32 + j];
        const float bxn = b_ih[64 + j];
        const float bhn = b_hh[64 + j];
#pragma unroll
        for (int i = 0; i < 8; ++i) {
            const int m  = i + mhi;
            const float rg = sigmoid_f(accR[ft][i] + br);
            const float zg = sigmoid_f(accZ[ft][i] + bz);
            const float ng = tanhf(accXN[ft][i] + bxn + rg * (accHN[ft][i] + bhn));
            const float hv = hbase[(size_t)m * 32 + j];
            ob[(size_t)m * 32 + j] = (1.0f - zg) * ng + zg * hv;
        }
    }
}

extern "C" void kernel_launch(void* const* d_in, const int* in_sizes, int n_in,
                              void* d_out, int out_size, void* d_ws, size_t ws_size,
                              hipStream_t stream) {
    (void)in_sizes; (void)n_in; (void)out_size; (void)d_ws; (void)ws_size;
    const float* pathf        = (const float*)d_in[0];
    const float* chanf        = (const float*)d_in[1];
    const float* W_ih         = (const float*)d_in[2];
    const float* W_hh         = (const float*)d_in[3];
    const float* b_ih         = (const float*)d_in[4];
    const float* b_hh         = (const float*)d_in[5];
    const int*   edge_path    = (const int*)d_in[6];
    const int*   edge_channel = (const int*)d_in[7];
    float* out = (float*)d_out;

    const int tiles  = (B_ * C_) / 16;   // 16384 waves
    const int blocks = tiles / 8;        // 8 waves (256 threads) per block
    gru_edge_fused_kernel<<<blocks, 256, 0, stream>>>(
        pathf, chanf, W_ih, W_hh, b_ih, b_hh, edge_path, edge_channel, out);
}